// GAT_60232621359631
// MI455X (gfx1250) — compile-verified
//
#include <hip/hip_runtime.h>
#include <hip/hip_bf16.h>

typedef float v2f __attribute__((ext_vector_type(2)));
typedef float v8f __attribute__((ext_vector_type(8)));

#define N_NODES 50000
#define N_EDGES 800000
#define N_FEAT  256
#define N_HID   32
#define N_HEADS 8
#define ALPHA   0.2f

// ---------------------------------------------------------------------------
// init: zero output, segmax = -inf, denom = 0 (graph-capture safe, no memset)
// ---------------------------------------------------------------------------
__global__ void gat_init(float* __restrict__ out,
                         float* __restrict__ segmax,
                         float* __restrict__ denom) {
    int i = blockIdx.x * blockDim.x + threadIdx.x;
    if (i < N_NODES * N_HEADS * N_HID) out[i] = 0.0f;
    if (i < N_NODES * N_HEADS) {
        segmax[i] = -__builtin_inff();
        denom[i]  = 0.0f;
    }
}

// ---------------------------------------------------------------------------
// GEMM: Wh[h][n][d] = sum_f h[n][f] * W[h][f][d]   via V_WMMA_F32_16X16X4_F32
// One block = 16-node tile; 8 waves, wave w computes head w (16x32 output).
// ---------------------------------------------------------------------------
__global__ __launch_bounds__(256) void gat_gemm(const float* __restrict__ hmat,
                                                const float* __restrict__ W,
                                                float* __restrict__ Wh) {
    __shared__ float sh[16 * N_FEAT];               // 16 KB fp32 h-tile

    const int node_base = blockIdx.x * 16;

    // 16 rows of h are contiguous in memory: straight 16KB block copy to LDS
    const float4* g4 = (const float4*)(hmat + (size_t)node_base * N_FEAT);
    float4* s4 = (float4*)sh;
    for (int i = threadIdx.x; i < (16 * N_FEAT) / 4; i += 256)
        s4[i] = g4[i];
    __syncthreads();

    const int lane = threadIdx.x & 31;
    const int head = threadIdx.x >> 5;              // 8 waves -> 8 heads
    const float* Wp = W + (size_t)head * N_FEAT * N_HID;

    const int m    = lane & 15;                     // M row (A) / N col (B)
    const int koff = (lane >> 4) << 1;              // lanes 16-31 hold K+2,K+3

    v8f c0 = {};                                    // N = 0..15
    v8f c1 = {};                                    // N = 16..31

    for (int k = 0; k < N_FEAT; k += 4) {
        v2f a, b0, b1;
        // A 16x4: lane<16 -> {K,K+1}, lane>=16 -> {K+2,K+3}, row m
        a[0] = sh[m * N_FEAT + k + koff];
        a[1] = sh[m * N_FEAT + k + koff + 1];
        // B 4x16: same K split across lane halves, col m; W row stride = 32
        b0[0] = Wp[(k + koff)     * N_HID + m];
        b0[1] = Wp[(k + koff + 1) * N_HID + m];
        b1[0] = Wp[(k + koff)     * N_HID + 16 + m];
        b1[1] = Wp[(k + koff + 1) * N_HID + 16 + m];
        c0 = __builtin_amdgcn_wmma_f32_16x16x4_f32(false, a, false, b0,
                                                   (short)0, c0, false, false);
        c1 = __builtin_amdgcn_wmma_f32_16x16x4_f32(false, a, false, b1,
                                                   (short)0, c1, false, false);
    }

    // C 16x16 f32: VGPR v, lanes 0-15 -> M=v, lanes 16-31 -> M=v+8
    float* op = Wh + ((size_t)head * N_NODES + node_base) * N_HID;
    const int mb = (lane >> 4) * 8;
#pragma unroll
    for (int v = 0; v < 8; ++v) {
        op[(mb + v) * N_HID + m]      = c0[v];
        op[(mb + v) * N_HID + 16 + m] = c1[v];
    }
}

// ---------------------------------------------------------------------------
// scores: s1[h][n] = Wh[h][n][:] . a[h][:32] ; s2 with a[h][32:64]
// one wave per (n,head), lane == d, shfl_xor reduction (wave32)
// ---------------------------------------------------------------------------
__global__ __launch_bounds__(256) void gat_score(const float* __restrict__ Wh,
                                                 const float* __restrict__ avec,
                                                 float* __restrict__ s1,
                                                 float* __restrict__ s2) {
    const int g    = (blockIdx.x * blockDim.x + threadIdx.x) >> 5;
    const int lane = threadIdx.x & 31;
    if (g >= N_NODES * N_HEADS) return;
    const int head = g & 7;
    const int n    = g >> 3;

    float wh = Wh[((size_t)head * N_NODES + n) * N_HID + lane];
    float p1 = wh * avec[head * 2 * N_HID + lane];
    float p2 = wh * avec[head * 2 * N_HID + N_HID + lane];
#pragma unroll
    for (int off = 16; off > 0; off >>= 1) {
        p1 += __shfl_xor(p1, off, 32);
        p2 += __shfl_xor(p2, off, 32);
    }
    if (lane == 0) {
        s1[head * N_NODES + n] = p1;
        s2[head * N_NODES + n] = p2;
    }
}

// float atomic max via integer atomics (monotone under sign-split encoding)
__device__ __forceinline__ void atomicMaxFloat(float* addr, float val) {
    if (val >= 0.0f)
        atomicMax((int*)addr, __float_as_int(val));
    else
        atomicMin((unsigned int*)addr, __float_as_uint(val));
}

// ---------------------------------------------------------------------------
// pass 1: e = leaky_relu(s1[src]+s2[dst]); segment max over dst
// grid: (edges/256, heads)
// ---------------------------------------------------------------------------
__global__ void gat_edge_max(const float* __restrict__ s1,
                             const float* __restrict__ s2,
                             const int* __restrict__ src,
                             const int* __restrict__ dst,
                             float* __restrict__ ebuf,
                             float* __restrict__ segmax) {
    const int e = blockIdx.x * blockDim.x + threadIdx.x;
    const int head = blockIdx.y;
    if (e >= N_EDGES) return;
    const int s = src[e], d = dst[e];
    float x = s1[head * N_NODES + s] + s2[head * N_NODES + d];
    float v = x > 0.0f ? x : ALPHA * x;
    ebuf[(size_t)head * N_EDGES + e] = v;
    atomicMaxFloat(&segmax[head * N_NODES + d], v);
}

// ---------------------------------------------------------------------------
// pass 2: ex = exp(e - segmax[dst]); segment sum -> denom
// ---------------------------------------------------------------------------
__global__ void gat_edge_exp(const int* __restrict__ dst,
                             const float* __restrict__ segmax,
                             float* __restrict__ ebuf,
                             float* __restrict__ denom) {
    const int e = blockIdx.x * blockDim.x + threadIdx.x;
    const int head = blockIdx.y;
    if (e >= N_EDGES) return;
    const int d = dst[e];
    float ex = __expf(ebuf[(size_t)head * N_EDGES + e] - segmax[head * N_NODES + d]);
    ebuf[(size_t)head * N_EDGES + e] = ex;
    atomicAdd(&denom[head * N_NODES + d], ex);
}

// ---------------------------------------------------------------------------
// pass 3: out[dst][h*32+d] += (ex/denom[dst]) * Wh[h][src][d]
// one wave per edge; lane == d (128B coalesced gather + atomic burst per head)
// ---------------------------------------------------------------------------
__global__ __launch_bounds__(256) void gat_message(const int* __restrict__ src,
                                                   const int* __restrict__ dst,
                                                   const float* __restrict__ ebuf,
                                                   const float* __restrict__ denom,
                                                   const float* __restrict__ Wh,
                                                   float* __restrict__ out) {
    const int e    = (blockIdx.x * blockDim.x + threadIdx.x) >> 5;
    const int lane = threadIdx.x & 31;
    if (e >= N_EDGES) return;
    const int s = src[e], d = dst[e];
    float* op = out + (size_t)d * (N_HEADS * N_HID);
#pragma unroll
    for (int head = 0; head < N_HEADS; ++head) {
        float attn = ebuf[(size_t)head * N_EDGES + e] / denom[head * N_NODES + d];
        float val  = attn * Wh[((size_t)head * N_NODES + s) * N_HID + lane];
        atomicAdd(op + head * N_HID + lane, val);
    }
}

// ---------------------------------------------------------------------------
extern "C" void kernel_launch(void* const* d_in, const int* in_sizes, int n_in,
                              void* d_out, int out_size, void* d_ws, size_t ws_size,
                              hipStream_t stream) {
    const float* h   = (const float*)d_in[0];
    const float* W   = (const float*)d_in[1];
    const float* a   = (const float*)d_in[2];
    const int*   src = (const int*)d_in[3];
    const int*   dst = (const int*)d_in[4];
    float* out = (float*)d_out;

    // workspace layout (floats)
    float* ws     = (float*)d_ws;
    float* Wh     = ws;                                        // H*N*D = 12.8M
    float* s1     = Wh + (size_t)N_HEADS * N_NODES * N_HID;    // H*N
    float* s2     = s1 + (size_t)N_HEADS * N_NODES;            // H*N
    float* segmax = s2 + (size_t)N_HEADS * N_NODES;            // H*N
    float* denom  = segmax + (size_t)N_HEADS * N_NODES;        // H*N
    float* ebuf   = denom + (size_t)N_HEADS * N_NODES;         // H*E = 6.4M

    gat_init<<<(N_NODES * N_HEADS * N_HID + 255) / 256, 256, 0, stream>>>(out, segmax, denom);

    gat_gemm<<<N_NODES / 16, 256, 0, stream>>>(h, W, Wh);

    gat_score<<<(N_NODES * N_HEADS * 32 + 255) / 256, 256, 0, stream>>>(Wh, a, s1, s2);

    dim3 eg((N_EDGES + 255) / 256, N_HEADS);
    gat_edge_max<<<eg, 256, 0, stream>>>(s1, s2, src, dst, ebuf, segmax);
    gat_edge_exp<<<eg, 256, 0, stream>>>(dst, segmax, ebuf, denom);

    gat_message<<<(N_EDGES * 32 + 255) / 256, 256, 0, stream>>>(src, dst, ebuf, denom, Wh, out);
}